// DCN_v1_14645838480159
// MI455X (gfx1250) — compile-verified
//
#include <hip/hip_runtime.h>
#include <hip/hip_bf16.h>
#include <math.h>
#include <stdint.h>

typedef __attribute__((ext_vector_type(2))) float v2f;
typedef __attribute__((ext_vector_type(8))) float v8f;

#define CIN   64
#define COUT  128
#define COFF  18
#define HH    128
#define WW    128
#define KK9   9
#define KTOT  (CIN * KK9)          // 576
#define OFF_ELEMS ((size_t)8 * COFF * HH * WW)   // 2,359,296

// ---------------------------------------------------------------------------
// Kernel 1: offset = conv3x3(x, w_off) + b_off, as a WMMA GEMM.
//   M = 18 (padded to 32 via zero-masked A rows), K = 576, N = pixels.
// One workgroup = one batch row, one 64-column half. 8 waves:
//   wave = (segment 0..3) x (M-tile 0..1). The 3x66x64 input tile is staged
//   straight into LDS with gfx1250 async global->LDS loads (ASYNCcnt),
//   bypassing the VGPR round-trip; only the zero halo uses ds_store.
// ---------------------------------------------------------------------------
__global__ __launch_bounds__(256) void dcn_offset_conv_wmma_kernel(
    const float* __restrict__ x, const float* __restrict__ w_off,
    const float* __restrict__ b_off, float* __restrict__ off_out)
{
    __shared__ float xs[CIN * 3 * 66];   // 12672 f32 = 50.7 KB
    __shared__ int   rofs[KTOT];         // row -> LDS offset (c*198 + ky*66 + kx)

    const int tid  = threadIdx.x;
    const int half = blockIdx.x & 1;
    const int h    = (blockIdx.x >> 1) % HH;
    const int b    = blockIdx.x / (2 * HH);
    const int w0   = half * 64;

    // Stage input tile: async global->LDS for in-bounds cells, ds_store 0 halo.
    const float* xb = x + (size_t)b * CIN * HH * WW;
    for (int i = tid; i < CIN * 198; i += 256) {
        const int c = i / 198, rem = i % 198;
        const int ry = rem / 66, cx = rem % 66;
        const int hy = h - 1 + ry, wx = w0 - 1 + cx;
        if (hy >= 0 && hy < HH && wx >= 0 && wx < WW) {
            const uint32_t ldsa = (uint32_t)(uintptr_t)&xs[i];
            const uint64_t ga   = (uint64_t)(uintptr_t)
                (xb + (size_t)c * HH * WW + (size_t)hy * WW + wx);
            asm volatile("global_load_async_to_lds_b32 %0, %1, off"
                         :: "v"(ldsa), "v"(ga) : "memory");
        } else {
            xs[i] = 0.f;
        }
    }

    // Overlap the row-offset table build with the async staging.
    for (int r = tid; r < KTOT; r += 256) {
        const int c = r / KK9, k9 = r % KK9;
        rofs[r] = c * 198 + (k9 / 3) * 66 + (k9 % 3);
    }

    asm volatile("s_wait_asynccnt 0" ::: "memory");   // drain async LDS writes
    __syncthreads();                                  // covers ds_store halo

    const int wid  = tid >> 5, lane = tid & 31;
    const int seg  = wid >> 1;              // 16-column segment within the half
    const int t    = wid & 1;               // M tile: channels [16t, 16t+16)
    const int n    = lane & 15;
    const int koff = (lane < 16) ? 0 : 2;
    const int m    = 16 * t + n;
    const float amask = (m < COFF) ? 1.f : 0.f;      // zero-pad rows 18..31
    const float* Arow = w_off + (size_t)min(m, COFF - 1) * KTOT;
    const int colbase = seg * 16 + n;

    v8f acc = {};
    for (int kk = 0; kk < KTOT; kk += 4) {
        v2f a, bm;
        a.x  = Arow[kk + koff] * amask;
        a.y  = Arow[kk + koff + 1] * amask;
        bm.x = xs[rofs[kk + koff] + colbase];
        bm.y = xs[rofs[kk + koff + 1] + colbase];
        acc = __builtin_amdgcn_wmma_f32_16x16x4_f32(
            false, a, false, bm, (short)0, acc, false, false);
    }

    const int mbase = 16 * t + ((lane < 16) ? 0 : 8);
    const int wcol  = w0 + seg * 16 + n;
#pragma unroll
    for (int i = 0; i < 8; ++i) {
        const int ch = mbase + i;
        if (ch < COFF)
            off_out[(((size_t)b * COFF + ch) * HH + h) * WW + wcol] = acc[i] + b_off[ch];
    }
}

// ---------------------------------------------------------------------------
// Kernel 2: deformable conv. One workgroup = one batch row, 32 pixels (two
// 16-pixel N-tiles), all 128 output channels. 256 threads = 8 wave32.
//   Phase 0: bilinear corner weights/indices for 9 taps x 32 pixels -> LDS
//   Phase 1: gather S[K=576][N=32] into LDS (L2-resident x)
//   Phase 2: per-wave 16-row M-tile, dual accumulators -> 2 WMMAs per A load
// ---------------------------------------------------------------------------
__global__ __launch_bounds__(256) void dcn_deform_wmma_kernel(
    const float* __restrict__ x, const float* __restrict__ off,
    const float* __restrict__ w_core, float* __restrict__ out)
{
    __shared__ float S[KTOT * 32];       // 18432 f32 = 73.7 KB, S[k][n] stride 32
    __shared__ float cw4[288][4];        // bilinear corner weights (0 if OOB)
    __shared__ int   ci4[288][4];        // clamped corner flat indices

    const int tid = threadIdx.x;
    const int bw  = blockIdx.x % (WW / 32);
    const int h   = (blockIdx.x / (WW / 32)) % HH;
    const int b   = blockIdx.x / ((WW / 32) * HH);
    const int w0  = bw * 32;

    // ---- Phase 0: sampling coordinates (9 taps x 32 pixels) ----------------
    for (int i = tid; i < 288; i += 256) {
        const int k = i / 32, p = i % 32;
        const int w = w0 + p;
        const float* ob = off + (((size_t)b * COFF + 2 * k) * HH + h) * WW + w;
        const float offy = ob[0];
        const float offx = ob[(size_t)HH * WW];
        const float py = (float)(h - 1 + k / 3) + offy;
        const float px = (float)(w - 1 + k % 3) + offx;
        const float y0f = floorf(py), x0f = floorf(px);
        const float dy = py - y0f, dx = px - x0f;
        const int y0 = (int)y0f, x0 = (int)x0f;
        const float wgt[4] = { (1.f - dy) * (1.f - dx), (1.f - dy) * dx,
                               dy * (1.f - dx),          dy * dx };
        const int yy[4] = { y0, y0, y0 + 1, y0 + 1 };
        const int xx[4] = { x0, x0 + 1, x0, x0 + 1 };
#pragma unroll
        for (int q = 0; q < 4; ++q) {
            const bool valid = (yy[q] >= 0) && (yy[q] < HH) && (xx[q] >= 0) && (xx[q] < WW);
            const int yc = min(max(yy[q], 0), HH - 1);
            const int xc = min(max(xx[q], 0), WW - 1);
            cw4[i][q] = valid ? wgt[q] : 0.f;
            ci4[i][q] = yc * WW + xc;
        }
    }
    __syncthreads();

    // ---- Phase 1: bilinear gather into S  (e = c*288 + k*32 + p) -----------
    const float* xb = x + (size_t)b * CIN * HH * WW;
    for (int e = tid; e < CIN * 288; e += 256) {   // 72 iters, uniform trip count
        const int c  = e / 288;
        const int kp = e % 288;
        const float* xc = xb + (size_t)c * HH * WW;
        S[e] = cw4[kp][0] * xc[ci4[kp][0]]
             + cw4[kp][1] * xc[ci4[kp][1]]
             + cw4[kp][2] * xc[ci4[kp][2]]
             + cw4[kp][3] * xc[ci4[kp][3]];
    }
    __syncthreads();

    // ---- Phase 2: GEMM, fp32 WMMA, 2 N-tiles per A fragment ----------------
    const int wid  = tid >> 5;          // wave id 0..7 -> M tile [16*wid, 16*wid+16)
    const int lane = tid & 31;
    const int n    = lane & 15;
    const int koff = (lane < 16) ? 0 : 2;
    const float* Wrow = w_core + (size_t)(wid * 16 + n) * KTOT;

    v8f acc0 = {}, acc1 = {};
    for (int kk = 0; kk < KTOT; kk += 4) {
        v2f a, b0, b1;
        a.x = Wrow[kk + koff];
        a.y = Wrow[kk + koff + 1];
        const int r0 = (kk + koff) * 32 + n;
        const int r1 = r0 + 32;
        b0.x = S[r0];      b0.y = S[r1];
        b1.x = S[r0 + 16]; b1.y = S[r1 + 16];
        acc0 = __builtin_amdgcn_wmma_f32_16x16x4_f32(
            false, a, false, b0, (short)0, acc0, false, false);
        acc1 = __builtin_amdgcn_wmma_f32_16x16x4_f32(
            false, a, false, b1, (short)0, acc1, false, false);
    }

    // ---- Write D: VGPR i -> M = i + (lane<16 ? 0 : 8), N = lane&15 ---------
    const int mbase = wid * 16 + ((lane < 16) ? 0 : 8);
#pragma unroll
    for (int i = 0; i < 8; ++i) {
        const int ch = mbase + i;
        float* op = out + (((size_t)b * COUT + ch) * HH + h) * WW + w0 + n;
        op[0]  = acc0[i];
        op[16] = acc1[i];
    }
}

// ---------------------------------------------------------------------------
extern "C" void kernel_launch(void* const* d_in, const int* in_sizes, int n_in,
                              void* d_out, int out_size, void* d_ws, size_t ws_size,
                              hipStream_t stream) {
    (void)in_sizes; (void)n_in; (void)out_size; (void)d_ws; (void)ws_size;
    const float* x      = (const float*)d_in[0];   // [8,64,128,128]
    const float* w_off  = (const float*)d_in[1];   // [18,64,3,3]
    const float* b_off  = (const float*)d_in[2];   // [18]
    const float* w_core = (const float*)d_in[3];   // [128,64,3,3]

    float* off_out = (float*)d_out;                // [8,18,128,128]
    float* out     = (float*)d_out + OFF_ELEMS;    // [8,128,128,128]

    // Kernel 1: 8 batches x 128 rows x 2 column-halves
    const int grid1 = 8 * HH * 2;
    dcn_offset_conv_wmma_kernel<<<grid1, 256, 0, stream>>>(x, w_off, b_off, off_out);

    // Kernel 2: 8 batches x 128 rows x 4 thirty-two-pixel segments
    const int grid2 = 8 * HH * (WW / 32);
    dcn_deform_wmma_kernel<<<grid2, 256, 0, stream>>>(x, off_out, w_core, out);
}